// GAT_68118181315267
// MI455X (gfx1250) — compile-verified
//
#include <hip/hip_runtime.h>
#include <hip/hip_bf16.h>

typedef __attribute__((ext_vector_type(2))) float v2f;
typedef __attribute__((ext_vector_type(8))) float v8f;

#define NEG_SLOPE 0.2f

// ---------------------------------------------------------------------------
// Zero-fill
// ---------------------------------------------------------------------------
__global__ void zero_f32(float* __restrict__ p, int n) {
    int i = blockIdx.x * blockDim.x + threadIdx.x;
    if (i < n) p[i] = 0.0f;
}

// ---------------------------------------------------------------------------
// FP32 WMMA GEMM: D[M,Ncols] = A[M,K] * B[K,Ncols]
// One wave computes a 16 x (16*NT) strip: NT accumulators, A loaded once per
// k-step and reused NT times; NT independent WMMA chains hide the D->D RAW
// hazard and load latency.  V_WMMA_F32_16X16X4_F32, K stepped by 4.
//   A (16x4 f32): lanes 0-15 rows M=0..15, vgpr pair = {K,K+1} (lanes 0-15)
//                 or {K+2,K+3} (lanes 16-31).  B (4x16) symmetric.
//   D: vgpr r -> row r + 8*half, lane lr -> column.
// ---------------------------------------------------------------------------
template<int KDIM, int NT>
__global__ void wmma_gemm_f32(const float* __restrict__ A,
                              const float* __restrict__ B,
                              float* __restrict__ D,
                              int M, int Ncols) {
    const int wave    = (blockIdx.x * blockDim.x + threadIdx.x) >> 5;
    const int ngroups = Ncols / (16 * NT);
    const int mt = wave / ngroups;
    const int ng = wave % ngroups;
    const int m0 = mt * 16;
    if (m0 >= M) return;                       // wave-uniform

    const int lane = threadIdx.x & 31;
    const int lr   = lane & 15;                // row (A) / col (B,D) in tile
    const int half = lane >> 4;                // K-pair select / D row-half

    int arow = m0 + lr;
    if (arow >= M) arow = M - 1;               // clamp (tail safety, uniform M)
    const float* __restrict__ ap = A + (size_t)arow * KDIM + 2 * half;
    const int    ncol0 = ng * (16 * NT) + lr;  // first column this lane touches

    v8f acc[NT] = {};
#pragma unroll 4
    for (int k = 0; k < KDIM; k += 4) {
        v2f a; a.x = ap[0]; a.y = ap[1];
        ap += 4;
        const float* __restrict__ bk = B + (size_t)(k + 2 * half) * Ncols + ncol0;
#pragma unroll
        for (int t = 0; t < NT; ++t) {
            v2f b; b.x = bk[t * 16]; b.y = bk[t * 16 + Ncols];
            acc[t] = __builtin_amdgcn_wmma_f32_16x16x4_f32(
                false, a, false, b, (short)0, acc[t], false, false);
        }
    }

    const int rbase = m0 + 8 * half;
    if (rbase + 7 < M) {                       // full tile: straight-line stores
#pragma unroll
        for (int t = 0; t < NT; ++t) {
            float* __restrict__ dp =
                D + (size_t)rbase * Ncols + ng * (16 * NT) + t * 16 + lr;
#pragma unroll
            for (int r = 0; r < 8; ++r) dp[(size_t)r * Ncols] = acc[t][r];
        }
    } else {                                   // ragged tail (never hit: M%16==0)
#pragma unroll
        for (int t = 0; t < NT; ++t) {
            float* __restrict__ dp =
                D + (size_t)rbase * Ncols + ng * (16 * NT) + t * 16 + lr;
#pragma unroll
            for (int r = 0; r < 8; ++r)
                if (rbase + r < M) dp[(size_t)r * Ncols] = acc[t][r];
        }
    }
}

// ---------------------------------------------------------------------------
// Attention coefficients: a_src[n,h] = sum_c H[n,h,c]*att_src[h,c]
// ---------------------------------------------------------------------------
template<int H, int C>
__global__ void attn_coef(const float* __restrict__ Hf,
                          const float* __restrict__ att_src,
                          const float* __restrict__ att_dst,
                          float* __restrict__ a_src,
                          float* __restrict__ a_dst,
                          int Nn) {
    int idx = blockIdx.x * blockDim.x + threadIdx.x;   // n*H + h
    if (idx >= Nn * H) return;
    const int h = idx % H;
    const int n = idx / H;
    const float* __restrict__ row = Hf + (size_t)n * (H * C) + h * C;
    const float* __restrict__ as  = att_src + h * C;
    const float* __restrict__ ad  = att_dst + h * C;
    float ss = 0.f, sd = 0.f;
#pragma unroll 8
    for (int c = 0; c < C; ++c) {
        const float v = row[c];
        ss += v * as[c];
        sd += v * ad[c];
    }
    a_src[idx] = ss;
    a_dst[idx] = sd;
}

// ---------------------------------------------------------------------------
// Edge pass 1: ex = exp(leaky_relu(a_src[s]+a_dst[d])); denom[d] += ex.
// Softmax without max-shift (exactly equivalent ratio; alpha is O(1) here so
// exp cannot overflow).  Edges e >= E are self-loops (s = d = e - E).
// ---------------------------------------------------------------------------
template<int H>
__global__ void edge_pass1(const int* __restrict__ srcI,
                           const int* __restrict__ dstI,
                           int E, int Nn,
                           const float* __restrict__ a_src,
                           const float* __restrict__ a_dst,
                           float* __restrict__ expb,
                           float* __restrict__ denom) {
    const int e = blockIdx.x * blockDim.x + threadIdx.x;
    const int Etot = E + Nn;
    if (e >= Etot) return;
    int s, d;
    if (e < E) { s = srcI[e]; d = dstI[e]; } else { s = d = e - E; }
#pragma unroll
    for (int h = 0; h < H; ++h) {
        float al = a_src[s * H + h] + a_dst[d * H + h];
        al = (al > 0.f) ? al : NEG_SLOPE * al;
        const float ex = __expf(al);
        expb[(size_t)e * H + h] = ex;
        atomicAdd(&denom[d * H + h], ex);   // global_atomic_add_f32, no return
    }
}

// ---------------------------------------------------------------------------
// Edge pass 2: one wave per edge; scatter weighted messages.
// agg[d, h*C + c] += Hf[s, h*C + c] * expb[e,h] / (denom[d,h] + 1e-16)
// ---------------------------------------------------------------------------
template<int H, int C>
__global__ void edge_pass2(const int* __restrict__ srcI,
                           const int* __restrict__ dstI,
                           int E, int Nn,
                           const float* __restrict__ Hf,
                           const float* __restrict__ expb,
                           const float* __restrict__ denom,
                           float* __restrict__ agg) {
    const int wave = (blockIdx.x * blockDim.x + threadIdx.x) >> 5;
    const int lane = threadIdx.x & 31;
    const int Etot = E + Nn;
    if (wave >= Etot) return;
    const int e = wave;
    int s, d;
    if (e < E) { s = srcI[e]; d = dstI[e]; } else { s = d = e - E; }
    constexpr int F = H * C;
    const float* __restrict__ hrow = Hf + (size_t)s * F;
    float* __restrict__ arow = agg + (size_t)d * F;
#pragma unroll
    for (int j = 0; j < F / 32; ++j) {
        const int i = j * 32 + lane;
        const int h = i / C;
        const float w = expb[(size_t)e * H + h] / (denom[d * H + h] + 1e-16f);
        atomicAdd(&arow[i], hrow[i] * w);
    }
}

// ---------------------------------------------------------------------------
// Epilogues
// ---------------------------------------------------------------------------
__global__ void bias_relu_inplace(float* __restrict__ p, const float* __restrict__ b,
                                  int Nn, int F) {
    int i = blockIdx.x * blockDim.x + threadIdx.x;
    if (i >= Nn * F) return;
    const float v = p[i] + b[i % F];
    p[i] = v > 0.f ? v : 0.f;
}

__global__ void bias_out(const float* __restrict__ p, const float* __restrict__ b,
                         float* __restrict__ o, int Nn, int F) {
    int i = blockIdx.x * blockDim.x + threadIdx.x;
    if (i >= Nn * F) return;
    o[i] = p[i] + b[i % F];
}

// ---------------------------------------------------------------------------
// Host launch
// ---------------------------------------------------------------------------
extern "C" void kernel_launch(void* const* d_in, const int* in_sizes, int n_in,
                              void* d_out, int out_size, void* d_ws, size_t ws_size,
                              hipStream_t stream) {
    const float* x        = (const float*)d_in[0];
    const int*   ei       = (const int*)d_in[1];   // [2,E] int32
    const float* W1       = (const float*)d_in[2];
    const float* att_src1 = (const float*)d_in[3];
    const float* att_dst1 = (const float*)d_in[4];
    const float* b1       = (const float*)d_in[5];
    const float* W2       = (const float*)d_in[6];
    const float* att_src2 = (const float*)d_in[7];
    const float* att_dst2 = (const float*)d_in[8];
    const float* b2       = (const float*)d_in[9];

    const int IN   = 256;
    const int Nn   = in_sizes[0] / IN;      // 20000
    const int E    = in_sizes[1] / 2;       // 640000
    const int Etot = E + Nn;
    const int* srcI = ei;
    const int* dstI = ei + E;

    // workspace layout (floats)
    float* ws    = (float*)d_ws;
    float* Hf    = ws;                                   // N*256 (L1) / N*64 (L2)
    float* agg1  = Hf   + (size_t)Nn * 256;              // N*256 -> L2 GEMM input
    float* asrc  = agg1 + (size_t)Nn * 256;              // N*8
    float* adst  = asrc + (size_t)Nn * 8;                // N*8
    float* denom = adst + (size_t)Nn * 8;                // N*8
    float* expb  = denom + (size_t)Nn * 8;               // (E+N)*8
    float* Hf2   = Hf;                                   // N*64 (dead Hf region)
    float* agg2  = Hf + (size_t)Nn * 64;                 // N*64 (dead Hf region)

    const int TB = 256;
    auto cdiv = [](int a, int b) { return (a + b - 1) / b; };

    // ============================ Layer 1 ============================
    zero_f32<<<cdiv(Nn * 256, TB), TB, 0, stream>>>(agg1, Nn * 256);
    zero_f32<<<cdiv(Nn * 8, TB), TB, 0, stream>>>(denom, Nn * 8);

    {   // Hf = x @ W1   (M=20000, K=256, Ncols=256) : 4 ngroups of 64 cols
        const int waves = cdiv(Nn, 16) * (256 / 64);
        wmma_gemm_f32<256, 4><<<cdiv(waves, 8), TB, 0, stream>>>(x, W1, Hf, Nn, 256);
    }
    attn_coef<8, 32><<<cdiv(Nn * 8, TB), TB, 0, stream>>>(Hf, att_src1, att_dst1,
                                                          asrc, adst, Nn);
    edge_pass1<8><<<cdiv(Etot, TB), TB, 0, stream>>>(srcI, dstI, E, Nn,
                                                     asrc, adst, expb, denom);
    edge_pass2<8, 32><<<cdiv(Etot, 8), TB, 0, stream>>>(srcI, dstI, E, Nn,
                                                        Hf, expb, denom, agg1);
    bias_relu_inplace<<<cdiv(Nn * 256, TB), TB, 0, stream>>>(agg1, b1, Nn, 256);

    // ============================ Layer 2 ============================
    zero_f32<<<cdiv(Nn * 64, TB), TB, 0, stream>>>(agg2, Nn * 64);
    zero_f32<<<cdiv(Nn, TB), TB, 0, stream>>>(denom, Nn);

    {   // Hf2 = relu(agg1) @ W2   (M=20000, K=256, Ncols=64) : 1 ngroup of 64
        const int waves = cdiv(Nn, 16) * (64 / 64);
        wmma_gemm_f32<256, 4><<<cdiv(waves, 8), TB, 0, stream>>>(agg1, W2, Hf2, Nn, 64);
    }
    attn_coef<1, 64><<<cdiv(Nn, TB), TB, 0, stream>>>(Hf2, att_src2, att_dst2,
                                                      asrc, adst, Nn);
    edge_pass1<1><<<cdiv(Etot, TB), TB, 0, stream>>>(srcI, dstI, E, Nn,
                                                     asrc, adst, expb, denom);
    edge_pass2<1, 64><<<cdiv(Etot, 8), TB, 0, stream>>>(srcI, dstI, E, Nn,
                                                        Hf2, expb, denom, agg2);
    bias_out<<<cdiv(Nn * 64, TB), TB, 0, stream>>>(agg2, b2, (float*)d_out, Nn, 64);
}